// DVDGN_53017076301932
// MI455X (gfx1250) — compile-verified
//
#include <hip/hip_runtime.h>
#include <math.h>

#define NN 50000
#define EE 300000
#define HH 64
#define PP 3
#define DIRS 3
#define LL 2
#define GG 50
#define FUS_IN 512
#define SLOPE 0.01f

typedef float v2f __attribute__((ext_vector_type(2)));
typedef float v8f __attribute__((ext_vector_type(8)));

// ---------------- WMMA core: one wave computes a 16x16 fp32 tile ----------------
// A (16 x K) is staged in LDS with row stride (K+1). B is [K,64] row-major in
// global (L2-resident weights), column window pre-offset by n0.
template <int K>
__device__ __forceinline__ v8f wmma_rowtile(const float* __restrict__ lds,
                                            const float* __restrict__ B,
                                            int lane) {
  v8f acc = {0.f, 0.f, 0.f, 0.f, 0.f, 0.f, 0.f, 0.f};
  const int m16 = lane & 15;
  const int half = lane >> 4;
  const int LDA = K + 1;
#pragma unroll 4
  for (int k0 = 0; k0 < K; k0 += 4) {
    const int ka = k0 + 2 * half;
    v2f a, b;
    a.x = lds[m16 * LDA + ka];
    a.y = lds[m16 * LDA + ka + 1];
    b.x = B[(size_t)ka * 64 + m16];
    b.y = B[(size_t)(ka + 1) * 64 + m16];
    acc = __builtin_amdgcn_wmma_f32_16x16x4_f32(false, a, false, b, (short)0,
                                                acc, false, false);
  }
  return acc;
}

// ---------------- GEMM: C[N,64] = A[N,64] @ B[64,64] + bias ----------------
__global__ void __launch_bounds__(128) gemm_a64(const float* __restrict__ A,
                                                const float* __restrict__ B,
                                                const float* __restrict__ bias,
                                                float* __restrict__ C) {
  __shared__ float lds[16 * 65];
  const int row0 = blockIdx.x * 16;
  const int tid = threadIdx.x;
  for (int idx = tid; idx < 16 * 64; idx += 128) {
    const int r = idx >> 6, c = idx & 63;
    lds[r * 65 + c] = A[(size_t)(row0 + r) * 64 + c];
  }
  __syncthreads();
  const int wave = tid >> 5, lane = tid & 31, n0 = wave * 16;
  v8f acc = wmma_rowtile<64>(lds, B + n0, lane);
  const int m16 = lane & 15, half = lane >> 4;
#pragma unroll
  for (int r = 0; r < 8; ++r) {
    const int m = r + 8 * half;
    const int n = n0 + m16;
    C[(size_t)(row0 + m) * 64 + n] = acc[r] + bias[n];
  }
}

// ---------------- Fusion GEMM: OUT += leaky(hcat @ fW + fb) * dw + hb ----------
// hcat columns: [0,64) = XT ; [64,512) = SC * deg_inv  (deg_inv folded here)
__global__ void __launch_bounds__(128) gemm_fus(
    const float* __restrict__ XT, const float* __restrict__ SC,
    const float* __restrict__ dinv, const float* __restrict__ fW,
    const float* __restrict__ fb, const float* __restrict__ dw,
    const float* __restrict__ hb, float* __restrict__ OUT) {
  __shared__ float lds[16 * 513];
  const int row0 = blockIdx.x * 16;
  const int tid = threadIdx.x;
  for (int idx = tid; idx < 16 * 512; idx += 128) {
    const int r = idx >> 9, c = idx & 511;
    const int row = row0 + r;
    float v;
    if (c < 64)
      v = XT[(size_t)row * 64 + c];
    else
      v = SC[(size_t)row * 448 + (c - 64)] * dinv[row];
    lds[r * 513 + c] = v;
  }
  __syncthreads();
  const int wave = tid >> 5, lane = tid & 31, n0 = wave * 16;
  v8f acc = wmma_rowtile<512>(lds, fW + n0, lane);
  const int m16 = lane & 15, half = lane >> 4;
#pragma unroll
  for (int r = 0; r < 8; ++r) {
    const int m = r + 8 * half;
    const int n = n0 + m16;
    float v = acc[r] + fb[n];
    v = v > 0.f ? v : v * SLOPE;
    const size_t o = (size_t)(row0 + m) * 64 + n;
    OUT[o] += v * dw[n] + hb[n];
  }
}

// ---------------- Gate GEMM: X = g*H + (1-g)*F + X, g = sigmoid([H|F]@W+b) ----
__global__ void __launch_bounds__(128) gemm_gate(
    const float* __restrict__ Hb, const float* __restrict__ Fb,
    const float* __restrict__ W, const float* __restrict__ gb,
    float* __restrict__ X) {
  __shared__ float lds[16 * 129];
  const int row0 = blockIdx.x * 16;
  const int tid = threadIdx.x;
  for (int idx = tid; idx < 16 * 128; idx += 128) {
    const int r = idx >> 7, c = idx & 127;
    const int row = row0 + r;
    float v = (c < 64) ? Hb[(size_t)row * 64 + c]
                       : Fb[(size_t)row * 64 + (c - 64)];
    lds[r * 129 + c] = v;
  }
  __syncthreads();
  const int wave = tid >> 5, lane = tid & 31, n0 = wave * 16;
  v8f acc = wmma_rowtile<128>(lds, W + n0, lane);
  const int m16 = lane & 15, half = lane >> 4;
#pragma unroll
  for (int r = 0; r < 8; ++r) {
    const int m = r + 8 * half;
    const int n = n0 + m16;
    const size_t o = (size_t)(row0 + m) * 64 + n;
    const float g = 1.f / (1.f + expf(-(acc[r] + gb[n])));
    X[o] = g * Hb[o] + (1.f - g) * Fb[o] + X[o];
  }
}

// ---------------- Edge scatter: one wave per edge, atomics into SC[N,448] -----
__global__ void __launch_bounds__(256) edge_scatter(
    const int* __restrict__ ei, const float* __restrict__ ew,
    const float* __restrict__ pe, const float* __restrict__ XT,
    float* __restrict__ SC, int p) {
  const int wid = (blockIdx.x * blockDim.x + threadIdx.x) >> 5;  // edge id
  const int lane = threadIdx.x & 31;
  if (wid >= EE) return;
  const int* rowp = ei + (size_t)p * 2 * EE;
  const int row = rowp[wid];
  const int col = rowp[EE + wid];
  const float w = ew[(size_t)p * EE + wid];
  float w2[6];
#pragma unroll
  for (int d = 0; d < 3; ++d) {
    const float diff = pe[(size_t)col * 3 + d] - pe[(size_t)row * 3 + d];
    w2[d] = diff > 0.f ? diff : 0.f;
    w2[3 + d] = diff < 0.f ? -diff : 0.f;
  }
  float* dst = SC + (size_t)row * 448;
#pragma unroll
  for (int t = 0; t < 2; ++t) {
    const int h = lane + 32 * t;
    const float xc = XT[(size_t)col * 64 + h];
    atomicAdd(dst + h, xc * w);
    float* dd = dst + 64 + h * 6;
#pragma unroll
    for (int d = 0; d < 6; ++d) atomicAdd(dd + d, xc * w2[d]);
  }
}

// ---------------- degree accumulation + inversion -----------------------------
__global__ void deg_accum(const int* __restrict__ ei,
                          const float* __restrict__ ew,
                          float* __restrict__ deg, int p) {
  const int e = blockIdx.x * blockDim.x + threadIdx.x;
  if (e >= EE) return;
  atomicAdd(deg + ei[(size_t)p * 2 * EE + e], ew[(size_t)p * EE + e]);
}

__global__ void deg_invert(float* __restrict__ deg, int n) {
  const int i = blockIdx.x * blockDim.x + threadIdx.x;
  if (i >= n) return;
  const float v = deg[i];
  deg[i] = (v == 0.f) ? 0.f : 1.f / v;
}

// ---------------- softmax over P for directional weights ----------------------
__global__ void softmax_dw(const float* __restrict__ Sd,
                           const float* __restrict__ Fd,
                           float* __restrict__ dwS, float* __restrict__ dwF) {
  const int id = blockIdx.x * blockDim.x + threadIdx.x;
  if (id >= 2 * LL * HH) return;
  const float* src = (id < LL * HH) ? Sd : Fd;
  float* dst = (id < LL * HH) ? dwS : dwF;
  const int lh = id & (LL * HH - 1);
  const int l = lh / HH, h = lh & (HH - 1);
  const float a0 = src[(l * PP + 0) * HH + h];
  const float a1 = src[(l * PP + 1) * HH + h];
  const float a2 = src[(l * PP + 2) * HH + h];
  const float mx = fmaxf(a0, fmaxf(a1, a2));
  const float e0 = expf(a0 - mx), e1 = expf(a1 - mx), e2 = expf(a2 - mx);
  const float s = e0 + e1 + e2;
  dst[(l * PP + 0) * HH + h] = e0 / s;
  dst[(l * PP + 1) * HH + h] = e1 / s;
  dst[(l * PP + 2) * HH + h] = e2 / s;
}

// ---------------- graph norm pieces -------------------------------------------
__global__ void cnt_accum(const int* __restrict__ batch,
                          float* __restrict__ cnt) {
  const int i = blockIdx.x * blockDim.x + threadIdx.x;
  if (i >= NN) return;
  atomicAdd(cnt + batch[i], 1.f);
}

__global__ void gn_sum(const float* __restrict__ x, const int* __restrict__ batch,
                       float* __restrict__ gsum) {
  const int idx = blockIdx.x * blockDim.x + threadIdx.x;
  if (idx >= NN * 64) return;
  const int i = idx >> 6, h = idx & 63;
  atomicAdd(gsum + batch[i] * 64 + h, x[idx]);
}

__global__ void gn_center(float* __restrict__ x, const int* __restrict__ batch,
                          const float* __restrict__ gsum,
                          const float* __restrict__ cnt,
                          const float* __restrict__ ms,
                          float* __restrict__ gsq) {
  const int idx = blockIdx.x * blockDim.x + threadIdx.x;
  if (idx >= NN * 64) return;
  const int i = idx >> 6, h = idx & 63;
  const int b = batch[i];
  const float c = fmaxf(cnt[b], 1.f);
  const float o = x[idx] - (gsum[b * 64 + h] / c) * ms[h];
  x[idx] = o;
  atomicAdd(gsq + b * 64 + h, o * o);
}

__global__ void gn_final(float* __restrict__ x, const int* __restrict__ batch,
                         const float* __restrict__ gsq,
                         const float* __restrict__ cnt,
                         const float* __restrict__ w,
                         const float* __restrict__ bb) {
  const int idx = blockIdx.x * blockDim.x + threadIdx.x;
  if (idx >= NN * 64) return;
  const int i = idx >> 6, h = idx & 63;
  const int b = batch[i];
  const float c = fmaxf(cnt[b], 1.f);
  const float var = gsq[b * 64 + h] / c;
  float y = x[idx] / sqrtf(var + 1e-5f) * w[h] + bb[h];
  y = y > 0.f ? y : y * SLOPE;
  x[idx] = y;
}

__global__ void readout_div(const float* __restrict__ gsum,
                            const float* __restrict__ cnt,
                            float* __restrict__ out) {
  const int idx = blockIdx.x * blockDim.x + threadIdx.x;
  if (idx >= GG * 64) return;
  out[idx] = gsum[idx] / fmaxf(cnt[idx >> 6], 1.f);
}

// =============================================================================
extern "C" void kernel_launch(void* const* d_in, const int* in_sizes, int n_in,
                              void* d_out, int out_size, void* d_ws,
                              size_t ws_size, hipStream_t stream) {
  (void)in_sizes; (void)n_in; (void)out_size; (void)ws_size;
  const float* in_x = (const float*)d_in[0];
  const float* pe = (const float*)d_in[1];
  const int* batch = (const int*)d_in[2];
  const int* uei = (const int*)d_in[3];
  const float* uew = (const float*)d_in[4];
  const int* fei = (const int*)d_in[5];
  const float* few = (const float*)d_in[6];
  const float* emb_W = (const float*)d_in[7];
  const float* emb_b = (const float*)d_in[8];
  const float* S_lin_W = (const float*)d_in[9];
  const float* S_lin_b = (const float*)d_in[10];
  const float* S_d = (const float*)d_in[11];
  const float* S_hb = (const float*)d_in[12];
  const float* S_fW = (const float*)d_in[13];
  const float* S_fb = (const float*)d_in[14];
  const float* F_lin_W = (const float*)d_in[15];
  const float* F_lin_b = (const float*)d_in[16];
  const float* F_d = (const float*)d_in[17];
  const float* F_hb = (const float*)d_in[18];
  const float* F_fW = (const float*)d_in[19];
  const float* F_fb = (const float*)d_in[20];
  const float* nS_w = (const float*)d_in[21];
  const float* nS_b = (const float*)d_in[22];
  const float* nS_m = (const float*)d_in[23];
  const float* nF_w = (const float*)d_in[24];
  const float* nF_b = (const float*)d_in[25];
  const float* nF_m = (const float*)d_in[26];
  const float* gate_W = (const float*)d_in[27];
  const float* gate_b = (const float*)d_in[28];
  float* out = (float*)d_out;

  // workspace carve-up (floats)
  float* ws = (float*)d_ws;
  size_t off = 0;
  float* X = ws + off;    off += (size_t)NN * 64;
  float* XT = ws + off;   off += (size_t)NN * 64;
  float* Hb = ws + off;   off += (size_t)NN * 64;
  float* Fb = ws + off;   off += (size_t)NN * 64;
  float* SC = ws + off;   off += (size_t)NN * 448;
  float* deg = ws + off;  off += (size_t)6 * NN;  // [u p0..2 | f p0..2], becomes deg_inv
  float* gsum = ws + off; off += GG * 64;
  float* gsq = ws + off;  off += GG * 64;
  float* cnt = ws + off;  off += GG;
  float* dwS = ws + off;  off += LL * PP * HH;
  float* dwF = ws + off;  off += LL * PP * HH;

  const int GEMM_BLKS = NN / 16;             // 3125
  const int ELEM_BLKS = (NN * 64) / 256;     // 12500
  const int EDGE_BLKS = EE / 8;              // 37500 (8 waves/block, 1 wave/edge)

  // ---- one-time precompute ----
  hipMemsetAsync(cnt, 0, GG * sizeof(float), stream);
  cnt_accum<<<(NN + 255) / 256, 256, 0, stream>>>(batch, cnt);
  hipMemsetAsync(deg, 0, (size_t)6 * NN * sizeof(float), stream);
  for (int p = 0; p < PP; ++p) {
    deg_accum<<<(EE + 255) / 256, 256, 0, stream>>>(uei, uew, deg + (size_t)p * NN, p);
    deg_accum<<<(EE + 255) / 256, 256, 0, stream>>>(fei, few, deg + (size_t)(3 + p) * NN, p);
  }
  deg_invert<<<(6 * NN + 255) / 256, 256, 0, stream>>>(deg, 6 * NN);
  softmax_dw<<<1, 256, 0, stream>>>(S_d, F_d, dwS, dwF);

  // ---- embedding ----
  gemm_a64<<<GEMM_BLKS, 128, 0, stream>>>(in_x, emb_W, emb_b, X);

  for (int i = 0; i < LL; ++i) {
    // ===== OPDMP (S) : input X -> Hb =====
    gemm_a64<<<GEMM_BLKS, 128, 0, stream>>>(X, S_lin_W + (size_t)i * HH * HH,
                                            S_lin_b + (size_t)i * HH, XT);
    hipMemsetAsync(Hb, 0, (size_t)NN * 64 * sizeof(float), stream);
    for (int p = 0; p < PP; ++p) {
      hipMemsetAsync(SC, 0, (size_t)NN * 448 * sizeof(float), stream);
      edge_scatter<<<EDGE_BLKS, 256, 0, stream>>>(uei, uew, pe, XT, SC, p);
      const size_t ip = (size_t)i * PP + p;
      gemm_fus<<<GEMM_BLKS, 128, 0, stream>>>(
          XT, SC, deg + (size_t)p * NN, S_fW + ip * FUS_IN * HH,
          S_fb + ip * HH, dwS + ip * HH, S_hb + ip * HH, Hb);
    }
    // graph norm + leaky on Hb
    hipMemsetAsync(gsum, 0, GG * 64 * sizeof(float), stream);
    hipMemsetAsync(gsq, 0, GG * 64 * sizeof(float), stream);
    gn_sum<<<ELEM_BLKS, 256, 0, stream>>>(Hb, batch, gsum);
    gn_center<<<ELEM_BLKS, 256, 0, stream>>>(Hb, batch, gsum, cnt,
                                             nS_m + (size_t)i * HH, gsq);
    gn_final<<<ELEM_BLKS, 256, 0, stream>>>(Hb, batch, gsq, cnt,
                                            nS_w + (size_t)i * HH,
                                            nS_b + (size_t)i * HH);

    // ===== OPDMP (F) : input Hb -> Fb =====
    gemm_a64<<<GEMM_BLKS, 128, 0, stream>>>(Hb, F_lin_W + (size_t)i * HH * HH,
                                            F_lin_b + (size_t)i * HH, XT);
    hipMemsetAsync(Fb, 0, (size_t)NN * 64 * sizeof(float), stream);
    for (int p = 0; p < PP; ++p) {
      hipMemsetAsync(SC, 0, (size_t)NN * 448 * sizeof(float), stream);
      edge_scatter<<<EDGE_BLKS, 256, 0, stream>>>(fei, few, pe, XT, SC, p);
      const size_t ip = (size_t)i * PP + p;
      gemm_fus<<<GEMM_BLKS, 128, 0, stream>>>(
          XT, SC, deg + (size_t)(3 + p) * NN, F_fW + ip * FUS_IN * HH,
          F_fb + ip * HH, dwF + ip * HH, F_hb + ip * HH, Fb);
    }
    // graph norm + leaky on Fb
    hipMemsetAsync(gsum, 0, GG * 64 * sizeof(float), stream);
    hipMemsetAsync(gsq, 0, GG * 64 * sizeof(float), stream);
    gn_sum<<<ELEM_BLKS, 256, 0, stream>>>(Fb, batch, gsum);
    gn_center<<<ELEM_BLKS, 256, 0, stream>>>(Fb, batch, gsum, cnt,
                                             nF_m + (size_t)i * HH, gsq);
    gn_final<<<ELEM_BLKS, 256, 0, stream>>>(Fb, batch, gsq, cnt,
                                            nF_w + (size_t)i * HH,
                                            nF_b + (size_t)i * HH);

    // ===== gate + residual: X = g*Hb + (1-g)*Fb + X =====
    gemm_gate<<<GEMM_BLKS, 128, 0, stream>>>(Hb, Fb, gate_W, gate_b, X);
  }

  // ---- readout: segment mean ----
  hipMemsetAsync(gsum, 0, GG * 64 * sizeof(float), stream);
  gn_sum<<<ELEM_BLKS, 256, 0, stream>>>(X, batch, gsum);
  readout_div<<<13, 256, 0, stream>>>(gsum, cnt, out);
}